// Isolate_loss_395136991424
// MI455X (gfx1250) — compile-verified
//
#include <hip/hip_runtime.h>
#include <hip/hip_bf16.h>

// CDNA5 / gfx1250 (wave32). WMMA f32 16x16x4 used for the weighted row-sum
// matvec s_b = w^T X_b; everything else is streaming reduction.

typedef float v2f __attribute__((ext_vector_type(2)));
typedef float v8f __attribute__((ext_vector_type(8)));

#define BATCH 128
#define NROWS 64
#define DCOLS 2048          // 64*32
#define EPSV  1e-12f

__device__ __forceinline__ float wave_reduce_sum(float v) {
    // wave32 butterfly reduction
    v += __shfl_xor(v, 16, 32);
    v += __shfl_xor(v, 8, 32);
    v += __shfl_xor(v, 4, 32);
    v += __shfl_xor(v, 2, 32);
    v += __shfl_xor(v, 1, 32);
    return v;
}

// Pass 1: one wave per row. ss = ||x_row||^2 ; w = 1/max(sqrt(ss), eps);
// diag = ss * w * w. 8192 rows -> 1024 blocks * 8 waves. HBM-rate streaming.
__global__ void rownorm_kernel(const float* __restrict__ X,
                               float* __restrict__ w_out,
                               float* __restrict__ diag_out) {
    const int wavesPerBlock = blockDim.x >> 5;
    const int row  = blockIdx.x * wavesPerBlock + (threadIdx.x >> 5);
    const int lane = threadIdx.x & 31;

    const float4* row4 = (const float4*)(X + (size_t)row * DCOLS);
    float ss = 0.0f;
#pragma unroll
    for (int i = 0; i < 16; ++i) {              // 2048 / (32 lanes * 4) = 16
        float4 v = row4[i * 32 + lane];
        ss = fmaf(v.x, v.x, ss);
        ss = fmaf(v.y, v.y, ss);
        ss = fmaf(v.z, v.z, ss);
        ss = fmaf(v.w, v.w, ss);
    }
    ss = wave_reduce_sum(ss);
    if (lane == 0) {
        float w = 1.0f / fmaxf(sqrtf(ss), EPSV);
        w_out[row]    = w;
        diag_out[row] = ss * w * w;             // == ||att_row||^2
    }
}

// Pass 2: s_b = sum_n w_n * X[b][n][:], via V_WMMA_F32_16X16X4_F32.
// grid = (128 batches, 8 column-chunks of 256 cols). 8 waves/block, each wave
// owns two 16-column tiles. A(16x4) = broadcast w rows; B(4x16) = X tile.
// ISA layouts (CDNA5 ISA): A lane<16 -> K=0,1 in v0,v1 (K=2,3 for lane>=16);
// B: N = lane&15, same K striping. C v0 lanes 0..15 = (M=0, N=lane) = s slice.
__global__ void matvec_wmma_kernel(const float* __restrict__ X,
                                   const float* __restrict__ w_in,
                                   float* __restrict__ partial) {
    const int b  = blockIdx.x;                   // batch
    const int cc = blockIdx.y;                   // column chunk (256 cols)

    __shared__ float s_w[NROWS];
    __shared__ float s_part[8];

    if (threadIdx.x < NROWS) s_w[threadIdx.x] = w_in[b * NROWS + threadIdx.x];
    __syncthreads();

    const int wv    = threadIdx.x >> 5;          // wave 0..7
    const int lane  = threadIdx.x & 31;
    const int n     = lane & 15;                 // column within tile
    const int khalf = lane >> 4;                 // K half selector

    const float* __restrict__ Xb = X + (size_t)b * NROWS * DCOLS;
    float acc = 0.0f;

#pragma unroll
    for (int tt = 0; tt < 2; ++tt) {
        const int d0 = cc * 256 + (wv * 2 + tt) * 16;
        v8f c = {};
#pragma unroll
        for (int g = 0; g < 16; ++g) {           // 16 row-groups x 4 rows = 64
            const int r0 = 4 * g + 2 * khalf;
            v2f a;
            a.x = s_w[r0];
            a.y = s_w[r0 + 1];
            v2f bb;
            bb.x = Xb[(size_t)r0 * DCOLS + d0 + n];
            bb.y = Xb[(size_t)(r0 + 1) * DCOLS + d0 + n];
            // D = A*B + C  (neg_a, A, neg_b, B, c_mod, C, reuse_a, reuse_b)
            c = __builtin_amdgcn_wmma_f32_16x16x4_f32(
                    false, a, false, bb, (short)0, c, false, false);
        }
        // c[0] lanes 0..15 hold s[d0+lane]; lanes 16..31 are duplicates.
        if (lane < 16) {
            float sv = c[0];
            acc = fmaf(sv, sv, acc);
        }
    }

    acc = wave_reduce_sum(acc);
    if (lane == 0) s_part[wv] = acc;
    __syncthreads();
    if (threadIdx.x == 0) {
        float t = 0.0f;
        for (int i = 0; i < 8; ++i) t += s_part[i];   // fixed order
        partial[b * 8 + cc] = t;
    }
}

// Pass 3: deterministic fixed-order/tree reduction to the scalar loss.
__global__ void finalize_kernel(const float* __restrict__ partial,  // 1024
                                const float* __restrict__ diag,     // 8192
                                float* __restrict__ out) {
    __shared__ float red[256];
    const int t = threadIdx.x;
    float s = 0.0f;
    for (int i = t; i < BATCH * 8; i += 256) s += partial[i];
    for (int i = t; i < BATCH * NROWS; i += 256) s -= diag[i];
    red[t] = s;
    __syncthreads();
    for (int stride = 128; stride > 0; stride >>= 1) {
        if (t < stride) red[t] += red[t + stride];
        __syncthreads();
    }
    if (t == 0) out[0] = red[0] * (1.0f / (float)BATCH);
}

extern "C" void kernel_launch(void* const* d_in, const int* in_sizes, int n_in,
                              void* d_out, int out_size, void* d_ws, size_t ws_size,
                              hipStream_t stream) {
    (void)in_sizes; (void)n_in; (void)out_size; (void)ws_size;
    const float* X = (const float*)d_in[0];      // (128, 64, 64, 32) f32
    float* ws       = (float*)d_ws;
    float* w_arr    = ws;                        // 8192 floats
    float* diag_arr = ws + BATCH * NROWS;        // 8192 floats
    float* part_arr = ws + 2 * BATCH * NROWS;    // 1024 floats

    // Pass 1: 8192 rows, one wave each (256 thr = 8 waves per block).
    rownorm_kernel<<<dim3((BATCH * NROWS) / 8), 256, 0, stream>>>(X, w_arr, diag_arr);
    // Pass 2: WMMA matvec + squared-sum partials (L2-resident reread).
    matvec_wmma_kernel<<<dim3(BATCH, 8), 256, 0, stream>>>(X, w_arr, part_arr);
    // Pass 3: deterministic scalar reduce.
    finalize_kernel<<<dim3(1), 256, 0, stream>>>(part_arr, diag_arr, (float*)d_out);
}